// SORJit_25305947308109
// MI455X (gfx1250) — compile-verified
//
#include <hip/hip_runtime.h>

#define N 256
#define OMEGA 1.5f

typedef __attribute__((ext_vector_type(2))) float v2f;
typedef __attribute__((ext_vector_type(8))) float v8f;

// ---------------------------------------------------------------------------
// Kernel A: per-batch triangular solve.
//   G1 = D + omega*L  (lower tri),  G = G1^-1 * (-(omega*U + (omega-1)*D))
//   f = G1^-1 * (omega*b)
// One block per batch, 288 threads: threads 0..255 each own one column of G,
// thread 256 owns the f column. Each thread's forward-substitution history
// lives in dynamic LDS (256 rows x 288 cols x 4B = 288KB of the 320KB WGP LDS).
// The A-row scanned in the inner loop is block-uniform -> scalar loads.
// ---------------------------------------------------------------------------
__global__ void sor_precompute(const float* __restrict__ A,
                               const float* __restrict__ b,
                               float* __restrict__ G,
                               float* __restrict__ f) {
    extern __shared__ float X[];               // [N][NC]
    const int NC = 288;
    const int bidx = blockIdx.x;
    const int c = threadIdx.x;                 // 0..287
    const float* Ab = A + (size_t)bidx * N * N;
    const float omega = OMEGA;

    for (int i = 0; i < N; ++i) {
        const float aii = Ab[i * N + i];
        float s;
        if (c < N) {
            s = (i < c)  ? -omega * Ab[i * N + c]
              : (i == c) ? -(omega - 1.0f) * aii
                         : 0.0f;
        } else {
            s = omega * b[bidx * N + i];
        }
        float acc = 0.0f;
        for (int j = 0; j < i; ++j) {
            acc = fmaf(Ab[i * N + j], X[j * NC + c], acc);
        }
        s -= omega * acc;
        X[i * NC + c] = s / aii;
    }
    __syncthreads();
    if (c < N) {
        float* Gb = G + (size_t)bidx * N * N;
        for (int i = 0; i < N; ++i) Gb[i * N + c] = X[i * NC + c];
        f[bidx * N + c] = X[c * NC + N];       // f[i] at column 256
    }
}

// ---------------------------------------------------------------------------
// Kernel B: 200 iterations of x <- G x + f per batch, err_t = ||xs - x_t||.
// One block (8 waves) per batch. Each wave owns rows [w*32, w*32+32) as two
// 16-row WMMA tiles; the GEMV is done with V_WMMA_F32_16X16X4_F32 where the
// B operand's 16 columns all hold the same 4-chunk of x (broadcast GEMV).
// G streams from L2 (128MB total, fits the 192MB L2 after iteration 0).
// Also emits err_0 at column 0 and xtol_b = rtol*||xs|| into workspace.
// ---------------------------------------------------------------------------
__global__ void sor_iterate(const float* __restrict__ G,
                            const float* __restrict__ f,
                            const float* __restrict__ xs,
                            const float* __restrict__ theta,
                            const float* __restrict__ rtol,
                            float* __restrict__ out,
                            float* __restrict__ xtol_ws,
                            int niter, int ostride) {
    __shared__ __align__(16) float xsh[N];
    __shared__ __align__(16) float xssh[N];
    __shared__ __align__(16) float fsh[N];
    __shared__ float redA[8];
    __shared__ float redB[8];

    const int bidx = blockIdx.x;
    const int tid  = threadIdx.x;              // 0..255
    const int lane = tid & 31;
    const int wv   = tid >> 5;                 // 0..7
    const int half = lane >> 4;                // 0|1
    const int m    = lane & 15;

    const float* Gb = G + (size_t)bidx * N * N;

    xsh[tid]  = theta[bidx * N + tid];
    xssh[tid] = xs[bidx * N + tid];
    fsh[tid]  = f[bidx * N + tid];
    __syncthreads();

    // err0 and xtol
    {
        float d  = xssh[tid] - xsh[tid];
        float e2 = d * d;
        float n2 = xssh[tid] * xssh[tid];
        for (int off = 16; off > 0; off >>= 1) {
            e2 += __shfl_xor(e2, off, 32);
            n2 += __shfl_xor(n2, off, 32);
        }
        if (lane == 0) { redA[wv] = e2; redB[wv] = n2; }
        __syncthreads();
        if (tid == 0) {
            float se = 0.f, sn = 0.f;
            for (int w = 0; w < 8; ++w) { se += redA[w]; sn += redB[w]; }
            out[(size_t)bidx * ostride + 0] = sqrtf(se);
            xtol_ws[bidx] = rtol[bidx] * sqrtf(sn);
        }
        __syncthreads();
    }

    const int r0 = wv * 32;                    // tile rows r0..r0+15
    const int r1 = r0 + 16;                    // tile rows r1..r1+15
    const float* arow0 = Gb + (size_t)(r0 + m) * N + 2 * half;
    const float* arow1 = Gb + (size_t)(r1 + m) * N + 2 * half;

    for (int t = 0; t < niter; ++t) {
        v8f c0 = {};
        v8f c1 = {};
        #pragma unroll 4
        for (int k = 0; k < N; k += 4) {
            v2f av0 = *(const v2f*)(arow0 + k);
            v2f av1 = *(const v2f*)(arow1 + k);
            v2f bv  = *(const v2f*)(&xsh[k + 2 * half]);  // all 16 cols = x chunk
            c0 = __builtin_amdgcn_wmma_f32_16x16x4_f32(
                     false, av0, false, bv, (short)0, c0, false, false);
            c1 = __builtin_amdgcn_wmma_f32_16x16x4_f32(
                     false, av1, false, bv, (short)0, c1, false, false);
        }
        __syncthreads();                       // all reads of xsh done
        if (m == 0) {                          // lanes 0 and 16: 8 rows each/tile
            const int rb0 = r0 + 8 * half;
            const int rb1 = r1 + 8 * half;
            #pragma unroll
            for (int v = 0; v < 8; ++v) {
                xsh[rb0 + v] = c0[v] + fsh[rb0 + v];
                xsh[rb1 + v] = c1[v] + fsh[rb1 + v];
            }
        }
        __syncthreads();                       // x_new visible
        float d  = xssh[tid] - xsh[tid];
        float e2 = d * d;
        for (int off = 16; off > 0; off >>= 1) e2 += __shfl_xor(e2, off, 32);
        if (lane == 0) redA[wv] = e2;
        __syncthreads();
        if (tid == 0) {
            float s = 0.f;
            for (int w = 0; w < 8; ++w) s += redA[w];
            out[(size_t)bidx * ostride + 1 + t] = sqrtf(s);
        }
    }
}

// ---------------------------------------------------------------------------
// Kernel C: K = min t in [0, niter) s.t. ALL batches have e_t <= xtol_b.
// out column j holds e_j exactly, so scan columns 0..niter-1.
// ---------------------------------------------------------------------------
__global__ void sor_findK(const float* __restrict__ out,
                          const float* __restrict__ xtol,
                          int bs, int ostride, int* __restrict__ Kws) {
    __shared__ int red[256];
    const int tid  = threadIdx.x;
    const int nit  = ostride - 1;
    int localK = nit;                          // nit => nothing zeroed
    for (int t = tid; t < nit; t += blockDim.x) {
        bool conv = true;
        for (int bb = 0; bb < bs; ++bb) {
            if (out[(size_t)bb * ostride + t] > xtol[bb]) { conv = false; break; }
        }
        if (conv && t < localK) localK = t;
    }
    red[tid] = localK;
    __syncthreads();
    for (int s = 128; s > 0; s >>= 1) {
        if (tid < s) red[tid] = (red[tid + s] < red[tid]) ? red[tid + s] : red[tid];
        __syncthreads();
    }
    if (tid == 0) *Kws = red[0];
}

// ---------------------------------------------------------------------------
// Kernel D: zero history columns j >= K+1 (column 0 = err0 never zeroed).
// ---------------------------------------------------------------------------
__global__ void sor_zero_tail(float* __restrict__ out,
                              const int* __restrict__ Kws, int ostride) {
    const int bidx = blockIdx.x;
    const int K = *Kws;
    for (int j = threadIdx.x; j < ostride; j += blockDim.x) {
        if (j >= K + 1) out[(size_t)bidx * ostride + j] = 0.0f;
    }
}

// ---------------------------------------------------------------------------
extern "C" void kernel_launch(void* const* d_in, const int* in_sizes, int n_in,
                              void* d_out, int out_size, void* d_ws, size_t ws_size,
                              hipStream_t stream) {
    const float* A     = (const float*)d_in[0];
    const float* b     = (const float*)d_in[1];
    const float* xs    = (const float*)d_in[2];
    const float* theta = (const float*)d_in[3];
    const float* rtol  = (const float*)d_in[4];
    float* out = (float*)d_out;

    const int bs      = in_sizes[0] / (N * N);   // 512
    const int ostride = out_size / bs;           // maxiter + 1 = 201
    const int niter   = ostride - 1;             // 200

    float* Gws    = (float*)d_ws;                         // bs*N*N floats (128MB)
    float* fws    = Gws + (size_t)bs * N * N;             // bs*N
    float* xtolws = fws + (size_t)bs * N;                 // bs
    int*   Kws    = (int*)(xtolws + bs);                  // 1

    const size_t ldsA = (size_t)N * 288 * sizeof(float);  // 288KB of 320KB WGP LDS

    sor_precompute<<<bs, 288, ldsA, stream>>>(A, b, Gws, fws);
    sor_iterate<<<bs, 256, 0, stream>>>(Gws, fws, xs, theta, rtol,
                                        out, xtolws, niter, ostride);
    sor_findK<<<1, 256, 0, stream>>>(out, xtolws, bs, ostride, Kws);
    sor_zero_tail<<<bs, 256, 0, stream>>>(out, Kws, ostride);
}